// KeypointDetectionSSC_5128190951771
// MI455X (gfx1250) — compile-verified
//
#include <hip/hip_runtime.h>
#include <stdint.h>

// Problem constants fixed by the harness's setup_inputs():
#define N_PTS   10000
#define NUM_RET 512
#define K_MIN   461      // round(512 * 0.9)
#define K_MAX   563      // round(512 * 1.1)
#define HI_INIT 960      // 3840 // int(sqrt(10000/512)) = 3840 // 4
#define R_CLAMP 511      // > max possible pad (960//2 = 480)

typedef __attribute__((ext_vector_type(4))) unsigned int u32x4;
typedef __attribute__((ext_vector_type(8))) int          i32x8;
typedef __attribute__((ext_vector_type(4))) int          i32x4;

__device__ __forceinline__ unsigned ballot32(bool p) {
#if __has_builtin(__builtin_amdgcn_ballot_w32)
  return __builtin_amdgcn_ballot_w32(p);
#else
  return (unsigned)__ballot(p);   // wave32: low 32 bits
#endif
}

// ---------------------------------------------------------------------------
// Kernel A: per-point NMS radius.
//   p_i    = max score at point i's pixel (collision-aware, matches .at[].max)
//   r_i    = min Chebyshev distance to any point with score > p_i (clamped 511)
//   canon  = 1 iff i is the first (highest-scoring) point at its pixel
// Point data staged into LDS once per block; scores via TDM DMA.
// ---------------------------------------------------------------------------
__global__ __launch_bounds__(256) void ssc_radius_kernel(
    const int* __restrict__ kp, const float* __restrict__ sc,
    int* __restrict__ rcanon)
{
  extern __shared__ char smem[];                       // 80 KB (of 320 KB/WGP)
  float* s_sc = (float*)smem;                          // [N_PTS] scores
  int*   s_xy = (int*)(smem + N_PTS * sizeof(float));  // [N_PTS] (y<<16)|x

#if __has_builtin(__builtin_amdgcn_tensor_load_to_lds)
  if (threadIdx.x == 0) {
    // Tensor DMA descriptor (D#): 1 row x 10000 elems, 4-byte data, no pad.
    unsigned lds_addr = (unsigned)(unsigned long long)(void*)s_sc; // LDS offset
    unsigned long long ga = (unsigned long long)sc;
    u32x4 g0;
    g0.x = 1u;                                     // count=1, user descriptor
    g0.y = lds_addr;                               // lds_addr (bytes)
    g0.z = (unsigned)(ga & 0xffffffffull);         // global_addr[31:0]
    g0.w = (unsigned)((ga >> 32) & 0x1ffffffull)   // global_addr[56:32]
         | 0x80000000u;                            // type=2 ("image")
    i32x8 g1;
    g1[0] = 0x00020000;                 // data_size=2 (4 bytes), mask=0
    g1[1] = (N_PTS & 0xffff) << 16;     // tensor_dim0[15:0]
    g1[2] = (N_PTS >> 16) | (1 << 16);  // tensor_dim0[31:16] | tensor_dim1=1
    g1[3] = N_PTS << 16;                // tile_dim0 = 10000
    g1[4] = 1;                          // tile_dim1 = 1, tile_dim2 = 0
    g1[5] = N_PTS;                      // tensor_dim0_stride[31:0]
    g1[6] = 0;
    g1[7] = 0;
    i32x4 gz4 = {0, 0, 0, 0};                       // groups 2/3 unused (<=2D)
    i32x8 gz8 = {0, 0, 0, 0, 0, 0, 0, 0};           // extra group (clang-23 sig)
    __builtin_amdgcn_tensor_load_to_lds(g0, g1, gz4, gz4, gz8, 0);
    __builtin_amdgcn_s_wait_tensorcnt(0);   // TENSORcnt is per-wave: wait here,
  }                                         // then workgroup barrier below.
#else
  for (int j = threadIdx.x; j < N_PTS; j += blockDim.x) s_sc[j] = sc[j];
#endif

  // Stage packed (y,x): one b128 load per keypoint row [0,0,y,x].
  const int4* kp4 = (const int4*)kp;
  for (int j = threadIdx.x; j < N_PTS; j += blockDim.x) {
    int4 row = kp4[j];
    s_xy[j] = (row.z << 16) | row.w;     // y in [0,2160), x in [0,3840)
  }
  __syncthreads();

  int i = blockIdx.x * blockDim.x + threadIdx.x;
  if (i >= N_PTS) return;

  int myxy = s_xy[i];
  float p  = s_sc[i];
  int canon = 1;
  // Pass 1: pixel max / canonical bit. Scores sorted descending, so only
  // j < i can raise the pixel max.
  for (int j = 0; j < i; ++j) {
    if (s_xy[j] == myxy) { canon = 0; p = fmaxf(p, s_sc[j]); }
  }
  // Pass 2: min Chebyshev distance to a strictly-higher-scored point.
  // Sorted-descending scores => early exit at first s_j <= p.
  int myy = myxy >> 16, myx = myxy & 0xffff;
  int r = R_CLAMP;
  for (int j = 0; j < N_PTS; ++j) {
    float sj = s_sc[j];
    if (sj <= p) break;
    int oxy = s_xy[j];
    int oy = oxy >> 16, ox = oxy & 0xffff;
    int dy = myy > oy ? myy - oy : oy - myy;
    int dx = myx > ox ? myx - ox : ox - myx;
    int d  = dy > dx ? dy : dx;
    r = d < r ? d : r;
  }
  rcanon[i] = (canon << 16) | r;
}

// ---------------------------------------------------------------------------
// Kernel B (single block): histogram of r over canonical points -> CDF,
// replicate the reference's host binary search exactly, then wave32
// ballot-compaction to build sel in reference order, write output.
// ---------------------------------------------------------------------------
__global__ __launch_bounds__(1024) void ssc_select_kernel(
    const int* __restrict__ kp, const float* __restrict__ sc,
    const int* __restrict__ rcanon, float* __restrict__ out)
{
  __shared__ int hist[R_CLAMP + 1];
  __shared__ int s_sel[NUM_RET];
  __shared__ int s_pad;

  for (int b = threadIdx.x; b <= R_CLAMP; b += blockDim.x) hist[b] = 0;
  __syncthreads();
  for (int i = threadIdx.x; i < N_PTS; i += blockDim.x) {
    int rc = rcanon[i];
    if (rc >> 16) atomicAdd(&hist[rc & 0xffff], 1);  // canonical pixels only
  }
  __syncthreads();

  if (threadIdx.x == 0) {
    int run = 0;                                  // hist -> inclusive CDF
    for (int b = 0; b <= R_CLAMP; ++b) { run += hist[b]; hist[b] = run; }
    const int P = run;                            // # distinct occupied pixels
    // Faithful replica of the reference binary search. nms_count(k) with the
    // even->odd adjustment has pad = k//2 in both cases; kept-pixel count at
    // pad is P - CDF(pad) (r_i > pad for the pixel's canonical point).
    int low = 1, high = HI_INIT, prev_k = -1, sel_pad = -1;
    for (;;) {
      int k = (low + high) >> 1;
      if (k == prev_k || low > high) break;
      int pad = k >> 1;
      int cnt = P - hist[pad < R_CLAMP ? pad : R_CLAMP];
      if (cnt >= K_MIN && cnt <= K_MAX) { sel_pad = pad; break; }
      if (cnt < K_MIN) high = k - 1; else low = k + 1;
      prev_k = k;
    }
    if (sel_pad < 0) sel_pad = (prev_k > 0 ? prev_k : 1) >> 1; // fallback
    s_pad = sel_pad;
  }
  __syncthreads();
  const int pad = s_pad;

  // Wave 0: ordered compaction. Pass 1 takes kept points ascending (this is
  // both flatnonzero order and the m>512 truncation); pass 2 fills with the
  // remaining (not-kept) points ascending, exactly as the reference does.
  if (threadIdx.x < 32) {
    int lane = threadIdx.x;
    int cnt = 0;
    for (int base = 0; base < N_PTS && cnt < NUM_RET; base += 32) {
      int i = base + lane;
      bool kept = (i < N_PTS) && ((rcanon[i] & 0xffff) > pad);
      unsigned mask = ballot32(kept);
      int pre = __popc(mask & ((1u << lane) - 1u));
      if (kept && cnt + pre < NUM_RET) s_sel[cnt + pre] = i;
      cnt += __popc(mask);
    }
    if (cnt < NUM_RET) {
      for (int base = 0; base < N_PTS && cnt < NUM_RET; base += 32) {
        int i = base + lane;
        bool take = (i < N_PTS) && !((rcanon[i] & 0xffff) > pad);
        unsigned mask = ballot32(take);
        int pre = __popc(mask & ((1u << lane) - 1u));
        if (take && cnt + pre < NUM_RET) s_sel[cnt + pre] = i;
        cnt += __popc(mask);
      }
    }
  }
  __syncthreads();

  // Output: keypoints[sel] (512x4) then scores[sel] (512), all as float.
  if (threadIdx.x < NUM_RET) {
    int t = threadIdx.x;
    int s = s_sel[t];
    int4 row = ((const int4*)kp)[s];
    out[4 * t + 0] = (float)row.x;
    out[4 * t + 1] = (float)row.y;
    out[4 * t + 2] = (float)row.z;
    out[4 * t + 3] = (float)row.w;
    out[4 * NUM_RET + t] = sc[s];
  }
}

extern "C" void kernel_launch(void* const* d_in, const int* in_sizes, int n_in,
                              void* d_out, int out_size, void* d_ws, size_t ws_size,
                              hipStream_t stream) {
  const int*   kp = (const int*)d_in[0];   // keypoints int32 [10000,4]
  const float* sc = (const float*)d_in[1]; // scores float32 [10000], descending
  float* out = (float*)d_out;              // 2560 floats
  int* rcanon = (int*)d_ws;                // 40 KB scratch

  const size_t lds_bytes = (size_t)N_PTS * 8;  // 80 KB of the 320 KB WGP LDS
  (void)hipFuncSetAttribute(reinterpret_cast<const void*>(ssc_radius_kernel),
                            hipFuncAttributeMaxDynamicSharedMemorySize,
                            (int)lds_bytes);

  dim3 gridA((N_PTS + 255) / 256);  // 40 blocks x 256 threads (8 waves each)
  ssc_radius_kernel<<<gridA, 256, lds_bytes, stream>>>(kp, sc, rcanon);
  ssc_select_kernel<<<1, 1024, 0, stream>>>(kp, sc, rcanon, out);
}